// ConvCNP1d_5025111737028
// MI455X (gfx1250) — compile-verified
//
#include <hip/hip_runtime.h>
#include <math.h>

#define BATCH 16
#define NC 2048
#define NT 2048
#define TGRID 2048
#define KEPS 1e-8f
#define MM_BLOCKS 64
// sqrt(0.5) * sqrt(log2(e)) : exp(-0.5*(d/ls)^2) == exp2(-(SCONST*d/ls)^2)
#define SCONST 0.84932180028801904272f

typedef float v2f __attribute__((ext_vector_type(2)));
typedef float v8f __attribute__((ext_vector_type(8)));

// ---------------------------------------------------------------------------
// Kernel 0a: partial min/max over xc and xt (64 blocks -> 64 partial pairs)
// ---------------------------------------------------------------------------
__global__ void minmax_part(const float* __restrict__ xc,
                            const float* __restrict__ xt,
                            int n, float* __restrict__ part) {
  __shared__ float smin[256];
  __shared__ float smax[256];
  int tid = threadIdx.x;
  float lo = 3.4e38f, hi = -3.4e38f;
  for (int i = blockIdx.x * 256 + tid; i < n; i += MM_BLOCKS * 256) {
    float a = xc[i]; lo = fminf(lo, a); hi = fmaxf(hi, a);
    float b = xt[i]; lo = fminf(lo, b); hi = fmaxf(hi, b);
  }
  smin[tid] = lo; smax[tid] = hi;
  __syncthreads();
  for (int s = 128; s > 0; s >>= 1) {
    if (tid < s) {
      smin[tid] = fminf(smin[tid], smin[tid + s]);
      smax[tid] = fmaxf(smax[tid], smax[tid + s]);
    }
    __syncthreads();
  }
  if (tid == 0) {
    part[blockIdx.x * 2 + 0] = smin[0];
    part[blockIdx.x * 2 + 1] = smax[0];
  }
}

// ---------------------------------------------------------------------------
// Kernel 0b: final min/max over the 64 partials
// ---------------------------------------------------------------------------
__global__ void minmax_final(const float* __restrict__ part,
                             float* __restrict__ mm) {
  __shared__ float smin[64];
  __shared__ float smax[64];
  int tid = threadIdx.x;  // 64 threads
  smin[tid] = part[tid * 2 + 0];
  smax[tid] = part[tid * 2 + 1];
  __syncthreads();
  for (int s = 32; s > 0; s >>= 1) {
    if (tid < s) {
      smin[tid] = fminf(smin[tid], smin[tid + s]);
      smax[tid] = fmaxf(smax[tid], smax[tid + s]);
    }
    __syncthreads();
  }
  if (tid == 0) { mm[0] = smin[0]; mm[1] = smax[0]; }
}

// ---------------------------------------------------------------------------
// Kernel 1: h = rbf(t, xc) @ [1, yc]  via V_WMMA_F32_16X16X4_F32.
// Pre-scaled coords fold ls, the 0.5, AND log2(e) into one constant so each
// kernel value is:  fma (diff) + mul (d*-d, free neg) + v_exp_f32 (native exp2).
// os is folded into the epilogue (linear contraction).
// A (16x4): fresh exp2 values.  B (4x16): col0=1, col1=yc (one FMA each).
// ---------------------------------------------------------------------------
__global__ __launch_bounds__(256) void rbf_gather_wmma(
    const float* __restrict__ xc, const float* __restrict__ yc,
    const float* __restrict__ ls_p, const float* __restrict__ os_p,
    const float* __restrict__ mm, float* __restrict__ hout) {
  int wave = (int)((blockIdx.x * blockDim.x + threadIdx.x) >> 5);
  int lane = threadIdx.x & 31;
  int b  = wave >> 7;            // TGRID/16 = 128 tiles per batch
  int t0 = (wave & 127) << 4;

  float lower = mm[0];
  float step  = (mm[1] - lower) / (float)(TGRID - 1);
  float ls = ls_p[0], os = os_p[0];
  float sc = SCONST / ls;

  int n    = lane & 15;           // A row == C column for this lane
  int koff = (lane >> 4) << 1;    // lanes 0-15 -> K{0,1}, lanes 16-31 -> K{2,3}
  float u1 = (lower + (float)(t0 + n) * step) * sc;
  float w0 = (n == 0) ? 1.0f : 0.0f;
  float w1 = (n == 1) ? 1.0f : 0.0f;

  const float* xcb = xc + b * NC;
  const float* ycb = yc + b * NC;

  v8f acc = {};
  for (int k = 0; k < NC; k += 4) {
    float2 xv = *(const float2*)(xcb + k + koff);
    float2 yv = *(const float2*)(ycb + k + koff);
    float d0 = fmaf(-sc, xv.x, u1);
    float d1 = fmaf(-sc, xv.y, u1);
    v2f a, bm;
    a.x = __builtin_amdgcn_exp2f(d0 * -d0);   // v_exp_f32, no log2e preamble
    a.y = __builtin_amdgcn_exp2f(d1 * -d1);
    bm.x = fmaf(w1, yv.x, w0);
    bm.y = fmaf(w1, yv.y, w0);
    acc = __builtin_amdgcn_wmma_f32_16x16x4_f32(
        false, a, false, bm, (short)0, acc, false, false);
  }
  if (n < 2) {                     // only C columns 0 (h0) and 1 (h1) are real
    int mbase = (lane >> 4) << 3;  // lanes 0-15: rows 0-7, lanes 16-31: rows 8-15
    float* dst = hout + (size_t)((b * TGRID + t0 + mbase) << 1) + n;
#pragma unroll
    for (int r = 0; r < 8; ++r) dst[2 * r] = os * acc[r];
  }
}

// ---------------------------------------------------------------------------
// Kernel 2: rep = [t, h0, h1/(h0+eps)] laid out (B, 3, T) for the convs.
// ---------------------------------------------------------------------------
__global__ void build_rep(const float* __restrict__ h,
                          const float* __restrict__ mm,
                          float* __restrict__ rep) {
  int i = blockIdx.x * blockDim.x + threadIdx.x;
  if (i >= BATCH * TGRID) return;
  int b = i / TGRID, j = i - b * TGRID;
  float lower = mm[0];
  float step  = (mm[1] - lower) / (float)(TGRID - 1);
  float h0 = h[2 * i];
  float h1 = h[2 * i + 1];
  rep[(b * 3 + 0) * TGRID + j] = lower + (float)j * step;
  rep[(b * 3 + 1) * TGRID + j] = h0;
  rep[(b * 3 + 2) * TGRID + j] = h1 / (h0 + KEPS);
}

// ---------------------------------------------------------------------------
// Kernel 3: generic conv1d (k=5, pad=2, NCH layout), optional ReLU.
// ---------------------------------------------------------------------------
__global__ void conv1d_k(const float* __restrict__ in, const float* __restrict__ w,
                         const float* __restrict__ bias, float* __restrict__ out,
                         int Cin, int Cout, int relu) {
  int idx = blockIdx.x * blockDim.x + threadIdx.x;
  int total = BATCH * Cout * TGRID;
  if (idx >= total) return;
  int x  = idx % TGRID;
  int co = (idx / TGRID) % Cout;
  int b  = idx / (TGRID * Cout);
  float s = bias[co];
  for (int ci = 0; ci < Cin; ++ci) {
    const float* ip = in + (size_t)(b * Cin + ci) * TGRID;
    const float* wp = w + (size_t)(co * Cin + ci) * 5;
#pragma unroll
    for (int k = 0; k < 5; ++k) {
      int xi = x + k - 2;
      if (xi >= 0 && xi < TGRID) s += wp[k] * ip[xi];
    }
  }
  out[idx] = relu ? fmaxf(s, 0.0f) : s;
}

// ---------------------------------------------------------------------------
// Kernel 4: g = [f_mu, softplus(f_sigma)] interleaved (B, T, 2).
// ---------------------------------------------------------------------------
__global__ void build_g(const float* __restrict__ c4, float* __restrict__ g) {
  int i = blockIdx.x * blockDim.x + threadIdx.x;
  if (i >= BATCH * TGRID) return;
  int b = i / TGRID, j = i - b * TGRID;
  float mu = c4[(b * 2 + 0) * TGRID + j];
  float sg = c4[(b * 2 + 1) * TGRID + j];
  float sp = (sg > 20.0f) ? sg : log1pf(__expf(sg));
  g[2 * i]     = mu;
  g[2 * i + 1] = sp;
}

// ---------------------------------------------------------------------------
// Kernel 5: out = rbf(xt, t) @ [f_mu, softplus(f_sigma)] via WMMA f32 16x16x4.
// ---------------------------------------------------------------------------
__global__ __launch_bounds__(256) void rbf_scatter_wmma(
    const float* __restrict__ xt, const float* __restrict__ g,
    const float* __restrict__ ls_p, const float* __restrict__ os_p,
    const float* __restrict__ mm, float* __restrict__ out) {
  int wave = (int)((blockIdx.x * blockDim.x + threadIdx.x) >> 5);
  int lane = threadIdx.x & 31;
  int b  = wave >> 7;            // NT/16 = 128 tiles per batch
  int i0 = (wave & 127) << 4;

  float lower = mm[0];
  float step  = (mm[1] - lower) / (float)(TGRID - 1);
  float ls = ls_p[0], os = os_p[0];
  float sc = SCONST / ls;

  int n    = lane & 15;
  int koff = (lane >> 4) << 1;
  float u1 = xt[(size_t)b * NT + i0 + n] * sc;
  float wsel  = (n < 2) ? 1.0f : 0.0f;
  int   gsel  = (n < 2) ? n : 0;   // clamp so loads stay in-bounds, zero via wsel
  float sstep = step * sc;
  float base  = lower * sc + (float)koff * sstep;
  const float* gb = g + (size_t)b * TGRID * 2;

  v8f acc = {};
  for (int k = 0; k < TGRID; k += 4) {
    float fk  = (float)k;
    float u2a = fmaf(fk, sstep, base);
    float u2b = u2a + sstep;
    float d0 = u1 - u2a;
    float d1 = u1 - u2b;
    v2f a, bm;
    a.x = __builtin_amdgcn_exp2f(d0 * -d0);
    a.y = __builtin_amdgcn_exp2f(d1 * -d1);
    float g0 = gb[((k + koff) << 1) + gsel];
    float g1 = gb[((k + koff + 1) << 1) + gsel];
    bm.x = wsel * g0;
    bm.y = wsel * g1;
    acc = __builtin_amdgcn_wmma_f32_16x16x4_f32(
        false, a, false, bm, (short)0, acc, false, false);
  }
  if (n < 2) {                   // col 0 = mu, col 1 = sigma
    int mbase = (lane >> 4) << 3;
    float* dst = out + (size_t)((b * NT + i0 + mbase) << 1) + n;
#pragma unroll
    for (int r = 0; r < 8; ++r) dst[2 * r] = os * acc[r];
  }
}

// ---------------------------------------------------------------------------
extern "C" void kernel_launch(void* const* d_in, const int* in_sizes, int n_in,
                              void* d_out, int out_size, void* d_ws, size_t ws_size,
                              hipStream_t stream) {
  const float* xc     = (const float*)d_in[0];
  const float* yc     = (const float*)d_in[1];
  const float* xt     = (const float*)d_in[2];
  const float* ls_psi = (const float*)d_in[3];
  const float* os_psi = (const float*)d_in[4];
  const float* ls_rho = (const float*)d_in[5];
  const float* os_rho = (const float*)d_in[6];
  const float* w1 = (const float*)d_in[7];
  const float* b1 = (const float*)d_in[8];
  const float* w2 = (const float*)d_in[9];
  const float* b2 = (const float*)d_in[10];
  const float* w3 = (const float*)d_in[11];
  const float* b3 = (const float*)d_in[12];
  const float* w4 = (const float*)d_in[13];
  const float* b4 = (const float*)d_in[14];
  float* out = (float*)d_out;
  float* ws  = (float*)d_ws;

  size_t o = 0;
  float* mm   = ws + o; o += 16;
  float* part = ws + o; o += MM_BLOCKS * 2;
  float* hraw = ws + o; o += (size_t)BATCH * TGRID * 2;
  float* rep  = ws + o; o += (size_t)BATCH * 3 * TGRID;
  float* c1   = ws + o; o += (size_t)BATCH * 16 * TGRID;
  float* c2   = ws + o; o += (size_t)BATCH * 32 * TGRID;
  float* c3   = ws + o; o += (size_t)BATCH * 16 * TGRID;
  float* c4b  = ws + o; o += (size_t)BATCH * 2 * TGRID;
  float* g    = ws + o; o += (size_t)BATCH * TGRID * 2 + 16;
  (void)ws_size; (void)in_sizes; (void)n_in; (void)out_size;

  int nbt = BATCH * TGRID;

  minmax_part<<<MM_BLOCKS, 256, 0, stream>>>(xc, xt, BATCH * NC, part);
  minmax_final<<<1, 64, 0, stream>>>(part, mm);

  // 2048 waves total = 256 blocks of 8 waves
  rbf_gather_wmma<<<(BATCH * (TGRID / 16)) / 8, 256, 0, stream>>>(
      xc, yc, ls_psi, os_psi, mm, hraw);

  build_rep<<<(nbt + 255) / 256, 256, 0, stream>>>(hraw, mm, rep);

  conv1d_k<<<(BATCH * 16 * TGRID + 255) / 256, 256, 0, stream>>>(rep, w1, b1, c1, 3, 16, 1);
  conv1d_k<<<(BATCH * 32 * TGRID + 255) / 256, 256, 0, stream>>>(c1, w2, b2, c2, 16, 32, 1);
  conv1d_k<<<(BATCH * 16 * TGRID + 255) / 256, 256, 0, stream>>>(c2, w3, b3, c3, 32, 16, 1);
  conv1d_k<<<(BATCH * 2  * TGRID + 255) / 256, 256, 0, stream>>>(c3, w4, b4, c4b, 16, 2, 0);

  build_g<<<(nbt + 255) / 256, 256, 0, stream>>>(c4b, g);

  rbf_scatter_wmma<<<(BATCH * (NT / 16)) / 8, 256, 0, stream>>>(
      xt, g, ls_rho, os_rho, mm, out);
}